// MaskRoIPoolHead_19593640804507
// MI455X (gfx1250) — compile-verified
//
#include <hip/hip_runtime.h>
#include <hip/hip_bf16.h>

typedef __attribute__((ext_vector_type(2))) float v2f;
typedef __attribute__((ext_vector_type(8))) float v8f;

#define SIDE 14
#define PTS  (SIDE*SIDE)   // 196
#define CCH  256
#define HH   256
#define WW   256
#define NBOX 256
#define MAXW 32            // max feature-window width (padded to mult of 4)
#define MAXH 32            // max feature-window height
#define NTHR 224           // 7 waves of 32

// LDS layout (floats):
//   featL : MAXH*MAXW*16   (window for 16 channels, layout [y][x][c16])
//   WXs   : MAXW*16        (x-interp weights, layout [x][gx16], zero-padded)
//   wy0s/wy1s : SIDE each  (y-interp weights per grid row)
//   y0rs  : SIDE ints      (window-relative y0 per grid row)
#define SM_FEAT 0
#define SM_WX   (MAXH*MAXW*16)
#define SM_WY0  (SM_WX + MAXW*16)
#define SM_WY1  (SM_WY0 + SIDE)
#define SM_Y0R  (SM_WY1 + SIDE)
#define SM_FLOATS (SM_Y0R + SIDE)

__global__ __launch_bounds__(NTHR)
void roi_sep_wmma_kernel(const float* __restrict__ feats,
                         const float* __restrict__ boxes,
                         float* __restrict__ out)
{
  extern __shared__ float smem[];
  float* featL = smem + SM_FEAT;
  float* WXs   = smem + SM_WX;
  float* wy0s  = smem + SM_WY0;
  float* wy1s  = smem + SM_WY1;
  int*   y0rs  = (int*)(smem + SM_Y0R);

  const int box  = blockIdx.x;        // b*NBOX + n, matches output row order
  const int b    = box >> 8;
  const int nb   = box & 255;
  const int tid  = threadIdx.x;
  const int lane = tid & 31;          // wave32
  const int wave = tid >> 5;          // 0..6
  const int lh   = lane >> 4;         // lane-half: selects K pair / M+8
  const int ln   = lane & 15;         // A: M row; B/D: N column

  const size_t boff = ((size_t)b * NBOX + nb) * 4;
  const float x1 = boxes[boff + 0];
  const float y1 = boxes[boff + 1];
  const float x2 = boxes[boff + 2];
  const float y2 = boxes[boff + 3];
  const float bw = x2 - x1, bh = y2 - y1;

  // window bounds from first/last grid samples (monotonic since bw,bh > 0)
  const float tA = 0.5f / (float)SIDE, tB = ((float)SIDE - 0.5f) / (float)SIDE;
  const float fx_lo = (x1 + tA * bw) * 0.25f - 0.5f;
  const float fx_hi = (x1 + tB * bw) * 0.25f - 0.5f;
  const float fy_lo = (y1 + tA * bh) * 0.25f - 0.5f;
  const float fy_hi = (y1 + tB * bh) * 0.25f - 0.5f;
  const int xlo = (int)floorf(fx_lo);
  const int ylo = (int)floorf(fy_lo);
  int Wwin = (int)floorf(fx_hi) + 2 - xlo;
  int Hwin = (int)floorf(fy_hi) + 2 - ylo;
  Wwin = Wwin < 2 ? 2 : (Wwin > MAXW ? MAXW : Wwin);
  Hwin = Hwin < 2 ? 2 : (Hwin > MAXH ? MAXH : Hwin);
  const int Wk = (Wwin + 3) & ~3;     // K row length, multiple of 4

  // ---- build interpolation tables (once per box) ----
  for (int i = tid; i < MAXW * 16; i += NTHR) WXs[i] = 0.0f;
  __syncthreads();
  if (tid < SIDE) {
    const float t  = ((float)tid + 0.5f) / (float)SIDE;
    const float fx = (x1 + t * bw) * 0.25f - 0.5f;
    const float fy = (y1 + t * bh) * 0.25f - 0.5f;
    const float x0f = floorf(fx), y0f = floorf(fy);
    const float wxf = fx - x0f,  wyf = fy - y0f;
    int xr = (int)x0f - xlo;
    xr = xr < 0 ? 0 : (xr > Wk - 2 ? Wk - 2 : xr);
    WXs[xr * 16 + tid]       = 1.0f - wxf;   // distinct column per gx: no race
    WXs[(xr + 1) * 16 + tid] = wxf;
    int yr = (int)y0f - ylo;
    yr = yr < 0 ? 0 : (yr > Hwin - 2 ? Hwin - 2 : yr);
    y0rs[tid] = yr;
    wy0s[tid] = 1.0f - wyf;
    wy1s[tid] = wyf;
  }

  // ---- loop over 16-channel chunks ----
  for (int chunk = 0; chunk < CCH / 16; ++chunk) {
    const int cbase = chunk * 16;
    __syncthreads();   // protect featL from previous iteration's readers

    // stage window: featL[y][x][c], zero-padded outside image / past Wwin
    {
      const int c    = tid & 15;
      const int ysub = tid >> 4;   // 0..13
      const float* fb = feats + ((size_t)(b * CCH + cbase + c)) * (size_t)(HH * WW);
      for (int y = ysub; y < Hwin; y += 14) {
        const int gyy = ylo + y;
        const bool yin = (gyy >= 0) && (gyy < HH);
        const float* rowp = fb + (size_t)(yin ? gyy : 0) * WW;
        float* dst = featL + (y * Wk) * 16 + c;
        for (int x = 0; x < Wk; ++x) {
          const int gxx = xlo + x;
          float v = 0.0f;
          if (yin && (x < Wwin) && gxx >= 0 && gxx < WW) v = rowp[gxx];
          dst[x * 16] = v;   // lanes c=0..15 -> consecutive: conflict-free
        }
      }
    }
    __syncthreads();

    // compute: each wave owns grid rows {wave, wave+7}
    for (int gy = wave; gy < SIDE; gy += 7) {
      const int r0 = y0rs[gy];
      v8f acc = {0.f, 0.f, 0.f, 0.f, 0.f, 0.f, 0.f, 0.f};
      #pragma unroll
      for (int half = 0; half < 2; ++half) {
        const int   row = r0 + half;
        const float wv  = half ? wy1s[gy] : wy0s[gy];
        const int   Kt  = Wk >> 2;
        for (int kt = 0; kt < Kt; ++kt) {
          // tile-local K index for this lane-half: k = 2*lh + {0,1}
          const int xq = 4 * kt + 2 * lh;
          const int fo = (row * Wk + xq) * 16;
          v2f a, bb;
          a.x  = featL[fo + ln];          // A[m][k]   m = ln
          a.y  = featL[fo + 16 + ln];     // A[m][k+1] (x+1 pixel)
          bb.x = WXs[xq * 16 + ln] * wv;  // B[k][n]   n = ln
          bb.y = WXs[(xq + 1) * 16 + ln] * wv;
          acc = __builtin_amdgcn_wmma_f32_16x16x4_f32(
                    false, a, false, bb, (short)0, acc, false, false);
        }
      }
      // D: VGPR i holds M = i + 8*lh, column N = ln (only gx<14 valid)
      if (ln < SIDE) {
        const size_t baseo =
            ((size_t)box * CCH + cbase + 8 * lh) * (size_t)PTS + gy * SIDE + ln;
        #pragma unroll
        for (int i = 0; i < 8; ++i)
          out[baseo + (size_t)i * PTS] = acc[i];
      }
    }
  }
}

extern "C" void kernel_launch(void* const* d_in, const int* in_sizes, int n_in,
                              void* d_out, int out_size, void* d_ws, size_t ws_size,
                              hipStream_t stream) {
  const float* feats = (const float*)d_in[0];  // [4,256,256,256] f32
  const float* boxes = (const float*)d_in[1];  // [4,256,4] f32
  float* out = (float*)d_out;                  // [1024,256,196] f32

  const int nblocks = 4 * NBOX;                // one block per box
  const size_t smem_bytes = (size_t)SM_FLOATS * sizeof(float);  // ~66 KB
  roi_sep_wmma_kernel<<<nblocks, NTHR, smem_bytes, stream>>>(feats, boxes, out);
}